// GCN_15204184228223
// MI455X (gfx1250) — compile-verified
//
#include <hip/hip_runtime.h>
#include <hip/hip_bf16.h>

// ---------------------------------------------------------------------------
// GCN forward for MI455X (gfx1250, wave32).
// Dense GEMMs (X@W1, h@W2, h@W3) use V_WMMA_F32_16X16X4_F32 (one wave per
// 16-row tile, K stepped by 4). Sparse aggregation is an L2-resident
// gather + atomic-scatter (h is only 6.4 MB, x is 51 MB -> both fit in the
// 192 MB L2), so it's latency bound, not HBM bound.
// ---------------------------------------------------------------------------

typedef float v2f __attribute__((ext_vector_type(2)));
typedef float v8f __attribute__((ext_vector_type(8)));

#define HIDDEN 16

// ---------------- degree / normalization ----------------

__global__ void deg_init_kernel(float* __restrict__ deg, int n) {
    int i = blockIdx.x * blockDim.x + threadIdx.x;
    if (i < n) deg[i] = 1.0f;  // self-loop
}

__global__ void deg_count_kernel(const int* __restrict__ dst, float* __restrict__ deg,
                                 int n_edges) {
    int e = blockIdx.x * blockDim.x + threadIdx.x;
    if (e < n_edges) atomicAdd(&deg[dst[e]], 1.0f);
}

__global__ void dinv_kernel(float* __restrict__ deg_dinv, int n) {
    int i = blockIdx.x * blockDim.x + threadIdx.x;
    if (i < n) deg_dinv[i] = rsqrtf(deg_dinv[i]);  // deg >= 1 always
}

// ---------------- dense GEMM via f32 WMMA ----------------
// H[n_rows,16] = X[n_rows,K] @ W[K,16].  One wave per 16-row tile.
// n_rows must be a multiple of 16 (100000 = 6250*16 here).
//
// A (16x4 f32) layout: lanes 0-15 hold rows M=lane with K = k,k+1 in v[0],v[1];
// lanes 16-31 hold the same rows with K = k+2,k+3.
// B (4x16) mirrors that K split; C/D is the 8-VGPR 16x16 layout.
__global__ void __launch_bounds__(256)
gemm_wmma_kernel(const float* __restrict__ X, const float* __restrict__ W,
                 float* __restrict__ H, int n_rows, int K) {
    int wave = (int)((blockIdx.x * blockDim.x + threadIdx.x) >> 5);
    int lane = threadIdx.x & 31;
    int row0 = wave * 16;
    if (row0 >= n_rows) return;          // wave-uniform: EXEC stays all-1s

    int half = lane >> 4;                // 0: K={k,k+1}, 1: K={k+2,k+3}
    int l16  = lane & 15;

    const float* xrow = X + (size_t)(row0 + l16) * K;
    v8f acc = {};
    for (int k = 0; k < K; k += 4) {
        int ka = k + half * 2;
        v2f a, b;
        a.x = xrow[ka];
        a.y = xrow[ka + 1];
        b.x = W[(size_t)ka * HIDDEN + l16];
        b.y = W[(size_t)(ka + 1) * HIDDEN + l16];
        acc = __builtin_amdgcn_wmma_f32_16x16x4_f32(
            /*neg_a=*/false, a, /*neg_b=*/false, b,
            /*c_mod=*/(short)0, acc, /*reuse_a=*/false, /*reuse_b=*/false);
    }
    // D: VGPR i -> row (i + 8*half), col l16
    float* out = H + (size_t)row0 * HIDDEN;
#pragma unroll
    for (int i = 0; i < 8; ++i)
        out[(size_t)(i + 8 * half) * HIDDEN + l16] = acc[i];
}

// ---------------- sparse aggregation ----------------

__global__ void zero_kernel(float* __restrict__ p, int n) {
    int i = blockIdx.x * blockDim.x + threadIdx.x;
    if (i < n) p[i] = 0.0f;
}

// 16 threads per edge (one per feature): coalesced 64B gather + 64B scatter.
__global__ void scatter_kernel(const int* __restrict__ src, const int* __restrict__ dst,
                               const float* __restrict__ dinv,
                               const float* __restrict__ hin,
                               float* __restrict__ agg, int n_edges) {
    int g = blockIdx.x * blockDim.x + threadIdx.x;
    int e = g >> 4;
    if (e >= n_edges) return;
    int f = g & 15;
    int s = src[e];
    int d = dst[e];
    float norm = dinv[s] * dinv[d];
    float v = hin[(size_t)s * HIDDEN + f] * norm;
    atomicAdd(&agg[(size_t)d * HIDDEN + f], v);
}

// agg[i,f] = (relu?)(agg[i,f] + dinv[i]^2 * hpre[i,f] + bias[f])
__global__ void finalize_kernel(const float* __restrict__ hpre,
                                const float* __restrict__ dinv,
                                const float* __restrict__ bias,
                                float* __restrict__ agg, int n_nodes, int do_relu) {
    int g = blockIdx.x * blockDim.x + threadIdx.x;
    int i = g >> 4;
    if (i >= n_nodes) return;
    int f = g & 15;
    float di = dinv[i];
    float v = agg[(size_t)i * HIDDEN + f] + di * di * hpre[(size_t)i * HIDDEN + f]
              + bias[f];
    if (do_relu) v = fmaxf(v, 0.0f);
    agg[(size_t)i * HIDDEN + f] = v;
}

// ---------------- pooling + classifier ----------------

__global__ void pool_kernel(const float* __restrict__ h, const int* __restrict__ batch,
                            float* __restrict__ pooled, int n_nodes) {
    int g = blockIdx.x * blockDim.x + threadIdx.x;
    int i = g >> 4;
    if (i >= n_nodes) return;
    int f = g & 15;
    atomicAdd(&pooled[(size_t)batch[i] * HIDDEN + f], h[(size_t)i * HIDDEN + f]);
}

__global__ void final_linear_kernel(const float* __restrict__ pooled,
                                    const float* __restrict__ Wlin,
                                    const float* __restrict__ blin,
                                    float* __restrict__ out,
                                    int n_graphs, int n_classes) {
    int g = blockIdx.x * blockDim.x + threadIdx.x;
    if (g >= n_graphs * n_classes) return;
    int gr = g / n_classes;
    int c  = g % n_classes;
    float acc = blin[c];
#pragma unroll
    for (int f = 0; f < HIDDEN; ++f)
        acc += pooled[(size_t)gr * HIDDEN + f] * Wlin[f * n_classes + c];
    out[g] = acc;
}

// ---------------------------------------------------------------------------

static inline size_t align_up(size_t v, size_t a) { return (v + a - 1) & ~(a - 1); }

extern "C" void kernel_launch(void* const* d_in, const int* in_sizes, int n_in,
                              void* d_out, int out_size, void* d_ws, size_t ws_size,
                              hipStream_t stream) {
    (void)n_in; (void)ws_size;

    const float* x    = (const float*)d_in[0];   // [N, 128]
    const int*   ei   = (const int*)d_in[1];     // [2, E] flat
    const int*   batch= (const int*)d_in[2];     // [N]
    const float* W1   = (const float*)d_in[3];   // [128, 16]
    const float* b1   = (const float*)d_in[4];
    const float* W2   = (const float*)d_in[5];   // [16, 16]
    const float* b2   = (const float*)d_in[6];
    const float* W3   = (const float*)d_in[7];   // [16, 16]
    const float* b3   = (const float*)d_in[8];
    const float* Wlin = (const float*)d_in[9];   // [16, 10]
    const float* blin = (const float*)d_in[10];  // [10]
    float* out = (float*)d_out;

    const int N_FEAT = 128;
    const int n_nodes = in_sizes[0] / N_FEAT;           // 100000
    const int n_edges = in_sizes[1] / 2;                // 3200000
    const int n_classes = in_sizes[10];                 // 10
    const int n_graphs  = out_size / n_classes;         // 512

    const int* src = ei;
    const int* dst = ei + n_edges;

    // ---- workspace layout ----
    char* ws = (char*)d_ws;
    size_t off = 0;
    float* dinv = (float*)(ws + off);  off = align_up(off + (size_t)n_nodes * 4, 256);
    float* bufA = (float*)(ws + off);  off = align_up(off + (size_t)n_nodes * HIDDEN * 4, 256);
    float* bufB = (float*)(ws + off);  off = align_up(off + (size_t)n_nodes * HIDDEN * 4, 256);
    float* pooled = (float*)(ws + off); off = align_up(off + (size_t)n_graphs * HIDDEN * 4, 256);

    const int T = 256;
    const int nodeBlocks  = (n_nodes + T - 1) / T;
    const int edgeBlocks  = (n_edges + T - 1) / T;
    const int nodeFBlocks = ((n_nodes * HIDDEN) + T - 1) / T;   // 16 threads / node
    const int edgeFBlocks = (int)(((long long)n_edges * HIDDEN + T - 1) / T);
    const int hElems      = n_nodes * HIDDEN;
    // GEMM: one wave per 16-row tile, 8 waves per 256-thread block
    const int tiles       = (n_nodes + 15) / 16;
    const int gemmBlocks  = (tiles + 7) / 8;

    // ---- degree normalization ----
    deg_init_kernel<<<nodeBlocks, T, 0, stream>>>(dinv, n_nodes);
    deg_count_kernel<<<edgeBlocks, T, 0, stream>>>(dst, dinv, n_edges);
    dinv_kernel<<<nodeBlocks, T, 0, stream>>>(dinv, n_nodes);

    // ---- layer 1: bufA = X@W1 ; bufB = A_norm bufA + b1, relu ----
    gemm_wmma_kernel<<<gemmBlocks, T, 0, stream>>>(x, W1, bufA, n_nodes, N_FEAT);
    zero_kernel<<<nodeFBlocks, T, 0, stream>>>(bufB, hElems);
    scatter_kernel<<<edgeFBlocks, T, 0, stream>>>(src, dst, dinv, bufA, bufB, n_edges);
    finalize_kernel<<<nodeFBlocks, T, 0, stream>>>(bufA, dinv, b1, bufB, n_nodes, 1);

    // ---- layer 2: bufA = bufB@W2 ; bufB = A_norm bufA + b2, relu ----
    gemm_wmma_kernel<<<gemmBlocks, T, 0, stream>>>(bufB, W2, bufA, n_nodes, HIDDEN);
    zero_kernel<<<nodeFBlocks, T, 0, stream>>>(bufB, hElems);
    scatter_kernel<<<edgeFBlocks, T, 0, stream>>>(src, dst, dinv, bufA, bufB, n_edges);
    finalize_kernel<<<nodeFBlocks, T, 0, stream>>>(bufA, dinv, b2, bufB, n_nodes, 1);

    // ---- layer 3: bufA = bufB@W3 ; bufB = A_norm bufA + b3 (no relu) ----
    gemm_wmma_kernel<<<gemmBlocks, T, 0, stream>>>(bufB, W3, bufA, n_nodes, HIDDEN);
    zero_kernel<<<nodeFBlocks, T, 0, stream>>>(bufB, hElems);
    scatter_kernel<<<edgeFBlocks, T, 0, stream>>>(src, dst, dinv, bufA, bufB, n_edges);
    finalize_kernel<<<nodeFBlocks, T, 0, stream>>>(bufA, dinv, b3, bufB, n_nodes, 0);

    // ---- global add pool + classifier ----
    zero_kernel<<<(n_graphs * HIDDEN + T - 1) / T, T, 0, stream>>>(pooled, n_graphs * HIDDEN);
    pool_kernel<<<nodeFBlocks, T, 0, stream>>>(bufB, batch, pooled, n_nodes);
    final_linear_kernel<<<(n_graphs * n_classes + T - 1) / T, T, 0, stream>>>(
        pooled, Wlin, blin, out, n_graphs, n_classes);
}